// SNN_1_2241972928691
// MI455X (gfx1250) — compile-verified
//
#include <hip/hip_runtime.h>

// CDNA5 / gfx1250. Wave32. f32 WMMA 16x16x4.
typedef float v2f __attribute__((ext_vector_type(2)));
typedef float v8f __attribute__((ext_vector_type(8)));

#define KDIM 60000      // flattened H*W
#define LDH  208        // padded N stride for H1 workspace (13 full 16-wide tiles)

// -----------------------------------------------------------------------------
// Kernel 1: H1[m, n] = sum_k X[m,k] * W1[n,k]
//   M = T*B = 1024, N = 200 (padded to 208), K = 60000
// Each block: one 32x16 output tile (two 16x16 WMMA tiles sharing the B
// fragment -> 1.5 loads per wmma, W1 traffic halved). 8 waves split K into
// contiguous 7500-element chunks (uniform/scalar bounds); LDS reduction.
// Grid = (13 N-tiles, 32 M-groups): adjacent blocks share the M rows so X is
// L2-reused across the 13 N-tiles; W1 (48MB) stays resident in the 192MB L2.
// Out-of-range W1 rows (n>=200) are clamped, not masked: they only write the
// H1 pad columns, which are never read.
// -----------------------------------------------------------------------------
__global__ __launch_bounds__(256)
void snn_gemm1(const float* __restrict__ X, const float* __restrict__ W1,
               float* __restrict__ H1)
{
    const int nb   = blockIdx.x * 16;     // N tile base (0..192)
    const int mb   = blockIdx.y * 32;     // M group base (0..992)
    const int tid  = threadIdx.x;
    // uniform wave index in an SGPR -> scalar loop control, SADDR-form loads
    const int wave = __builtin_amdgcn_readfirstlane(tid >> 5);
    const int lane = tid & 31;
    const int half = lane >> 4;           // 0: K={0,1}, 1: K={2,3} per ISA A layout
    const int l15  = lane & 15;

    // A fragments: rows mb+l15 and mb+16+l15, k offset half*2
    const float* aptr0 = X + (size_t)(mb + l15) * KDIM + (half << 1);
    const float* aptr1 = aptr0 + (size_t)16 * KDIM;   // +3.84MB: fits imm24 offset

    // B fragment: W1 row nb+l15 (clamped into bounds), same k offset
    int wrow = nb + l15;
    if (wrow > 199) wrow = 199;
    const float* bptr = W1 + (size_t)wrow * KDIM + (half << 1);

    // contiguous K chunk per wave: 60000 / 8 = 7500 -> 1875 steps = 375 * 5
    const int k0   = wave * (KDIM / 8);
    const int kend = k0 + (KDIM / 8);

    v8f c0 = {};
    v8f c1 = {};
    #pragma unroll 5
    for (int k = k0; k < kend; k += 4) {
        v2f b  = *(const v2f*)(bptr  + k);
        v2f a0 = *(const v2f*)(aptr0 + k);
        v2f a1 = *(const v2f*)(aptr1 + k);
        c0 = __builtin_amdgcn_wmma_f32_16x16x4_f32(false, a0, false, b,
                                                   (short)0, c0, false, false);
        c1 = __builtin_amdgcn_wmma_f32_16x16x4_f32(false, a1, false, b,
                                                   (short)0, c1, false, false);
    }

    // cross-wave reduction of both 16x16 f32 tiles through LDS (16 KB)
    __shared__ float red[2][8][256];
    #pragma unroll
    for (int r = 0; r < 8; ++r) {
        red[0][wave][r * 32 + lane] = c0[r];
        red[1][wave][r * 32 + lane] = c1[r];
    }
    __syncthreads();

    // tid -> (vgpr r, lane) -> (m, n) per 16x16 f32 C/D layout:
    //   m = r + (lane>=16 ? 8 : 0), n = lane & 15
    const int r2    = tid >> 5;
    const int lane2 = tid & 31;
    const int mloc  = r2 + ((lane2 >> 4) << 3);
    const int n     = lane2 & 15;
    #pragma unroll
    for (int tile = 0; tile < 2; ++tile) {
        float sum = 0.0f;
        #pragma unroll
        for (int w = 0; w < 8; ++w) sum += red[tile][w][tid];
        H1[(size_t)(mb + tile * 16 + mloc) * LDH + nb + n] = sum;
    }
}

// -----------------------------------------------------------------------------
// Kernel 2: sequential LIF recurrence over t=0..15 (single workgroup).
//   LIF1 on H1 -> s1 ; h2 = s1 @ W2^T via WMMA (fused LIF2) ; h3 scalar ; LIF3.
// State + spike buffers live in the workspace (L2-hot, tiny traffic per step).
// W2 rows >= 50 are clamped, not masked: those columns are dropped by the n < 50
// guard and never reach V2/S2.
// -----------------------------------------------------------------------------
__global__ __launch_bounds__(256)
void snn_lif_chain(const float* __restrict__ H1, const float* __restrict__ W2,
                   const float* __restrict__ W3, float* __restrict__ out,
                   float* __restrict__ V1, float* __restrict__ S1,
                   float* __restrict__ V2, float* __restrict__ S2,
                   float* __restrict__ V3)
{
    const int tid  = threadIdx.x;
    const int wave = __builtin_amdgcn_readfirstlane(tid >> 5);
    const int lane = tid & 31;
    const int half = lane >> 4;
    const int l15  = lane & 15;

    // deterministic state init every launch
    for (int i = tid; i < 12800; i += 256) V1[i] = 0.0f;
    for (int i = tid; i < 4096;  i += 256) V2[i] = 0.0f;
    if (tid < 128) V3[tid] = 0.0f;
    __threadfence();
    __syncthreads();

    for (int t = 0; t < 16; ++t) {
        // ---- Layer 1 LIF over [64 x 200] ----
        for (int i = tid; i < 12800; i += 256) {
            const int b = i / 200;
            const int j = i - b * 200;
            const float h = H1[(size_t)(t * 64 + b) * LDH + j];
            float v = V1[i];
            v += (h - v) * 0.5f;                       // tau = 2
            const float s = (v >= 1.0f) ? 1.0f : 0.0f; // v_th = 1, hard reset
            V1[i] = v * (1.0f - s);
            S1[i] = s;
        }
        __threadfence();
        __syncthreads();

        // ---- Layer 2: [64,200] x [200,50] via WMMA, 16 tiles / 8 waves ----
        #pragma unroll
        for (int pass = 0; pass < 2; ++pass) {
            const int tt = wave + pass * 8;   // 0..15
            const int mt = tt >> 2;           // 0..3 (M=64)
            const int nt = tt & 3;            // 0..3 (N padded 50->64)

            const float* ap = S1 + (size_t)(mt * 16 + l15) * 200 + (half << 1);
            int wr = nt * 16 + l15;
            if (wr > 49) wr = 49;
            const float* bp = W2 + (size_t)wr * 200 + (half << 1);

            v8f c = {};
            #pragma unroll 5
            for (int k = 0; k < 200; k += 4) {
                v2f a = *(const v2f*)(ap + k);
                v2f b = *(const v2f*)(bp + k);
                c = __builtin_amdgcn_wmma_f32_16x16x4_f32(false, a, false, b,
                                                          (short)0, c, false, false);
            }

            // fused LIF2 straight off the accumulator (C/D lane layout)
            #pragma unroll
            for (int r = 0; r < 8; ++r) {
                const int m = mt * 16 + r + (half << 3);
                const int n = nt * 16 + l15;
                if (n < 50) {
                    float v = V2[m * 64 + n];
                    v += (c[r] - v) * 0.5f;
                    const float s = (v >= 1.0f) ? 1.0f : 0.0f;
                    V2[m * 64 + n] = v * (1.0f - s);
                    S2[m * 64 + n] = s;
                }
            }
        }
        __threadfence();
        __syncthreads();

        // ---- Layer 3: [64,50] x [50,2] scalar + LIF3 -> output spikes ----
        if (tid < 128) {
            const int b = tid >> 1;
            const int o = tid & 1;
            float acc = 0.0f;
            #pragma unroll 10
            for (int k = 0; k < 50; ++k)
                acc += S2[b * 64 + k] * W3[o * 50 + k];
            float v = V3[tid];
            v += (acc - v) * 0.5f;
            const float s = (v >= 1.0f) ? 1.0f : 0.0f;
            V3[tid] = v * (1.0f - s);
            out[t * 128 + tid] = s;                    // [T,B,2] flat
        }
        __threadfence();
        __syncthreads();
    }
}

// -----------------------------------------------------------------------------
extern "C" void kernel_launch(void* const* d_in, const int* in_sizes, int n_in,
                              void* d_out, int out_size, void* d_ws, size_t ws_size,
                              hipStream_t stream)
{
    const float* x  = (const float*)d_in[0];  // [16,64,150,400] f32
    const float* W1 = (const float*)d_in[1];  // [200,60000]
    const float* W2 = (const float*)d_in[2];  // [50,200]
    const float* W3 = (const float*)d_in[3];  // [2,50]
    float* out = (float*)d_out;               // [16,64,2] f32

    float* ws = (float*)d_ws;                 // ~988 KB used
    float* H1 = ws;                           // 1024*208
    float* V1 = H1 + 1024 * LDH;              // 12800
    float* S1 = V1 + 12800;                   // 12800
    float* V2 = S1 + 12800;                   // 4096
    float* S2 = V2 + 4096;                    // 4096
    float* V3 = S2 + 4096;                    // 128

    dim3 grid1(13, 32);                       // N-tiles fastest => X L2 reuse
    snn_gemm1<<<grid1, 256, 0, stream>>>(x, W1, H1);
    snn_lif_chain<<<1, 256, 0, stream>>>(H1, W2, W3, out, V1, S1, V2, S2, V3);
}